// GACNet_12077448037165
// MI455X (gfx1250) — compile-verified
//
#include <hip/hip_runtime.h>
#include <hip/hip_bf16.h>

// ---------------------------------------------------------------------------
// GACNet set-abstraction layer for MI455X (gfx1250, wave32).
//   FPS -> ball query -> grouped features -> 3x (GEMM+BN+ReLU via
//   V_WMMA_F32_16X16X4_F32, pre-packed branch-free B-fragments) ->
//   graph-attention (WMMA e-matmul + in-register softmax) -> pooled output.
// Workspace requirement: ~556 MB in d_ws.
// ---------------------------------------------------------------------------

typedef float v2f __attribute__((ext_vector_type(2)));
typedef float v8f __attribute__((ext_vector_type(8)));

#define BATCH    16
#define NPTS     4096
#define NPOINT   1024
#define NSAMPLE  32
#define RADIUS2  0.04f
#define LRELU_A  0.2f
#define BN_EPS   1e-5f

#define NP_ROWS  (BATCH * NPOINT * NSAMPLE)   // 524288 grouped rows
#define FPS_ROWS (BATCH * NPOINT)             // 16384 fps rows
#define TOT_ROWS (NP_ROWS + FPS_ROWS)         // 540672 (divisible by 16)

// ---------------------------------------------------------------------------
// 1. Farthest point sampling: one block per batch, dist[] in LDS,
//    two-level wave32 shuffle argmax (first-index tie-break like jnp.argmax).
// ---------------------------------------------------------------------------
__global__ __launch_bounds__(1024) void fps_kernel(
    const float* __restrict__ xyz,       // (B,3,N)
    int*  __restrict__ fps_idx,          // (B,NPOINT)
    float* __restrict__ new_xyz,         // (B*NPOINT,3)
    float* __restrict__ out_newxyzT)     // d_out head: (B,3,NPOINT)
{
    const int b    = blockIdx.x;
    const int tid  = threadIdx.x;
    const int lane = tid & 31;
    const int wv   = tid >> 5;
    const float* xb = xyz + (size_t)b * 3 * NPTS;

    __shared__ float dist[NPTS];
    __shared__ float sv[32];
    __shared__ int   si[32];
    __shared__ int   sfar;

    for (int i = tid; i < NPTS; i += 1024) dist[i] = 1e10f;
    __syncthreads();

    int far = 0;
    for (int it = 0; it < NPOINT; ++it) {
        const float cx = xb[far];
        const float cy = xb[NPTS + far];
        const float cz = xb[2 * NPTS + far];
        if (tid == 0) {
            fps_idx[b * NPOINT + it] = far;
            new_xyz[(size_t)(b * NPOINT + it) * 3 + 0] = cx;
            new_xyz[(size_t)(b * NPOINT + it) * 3 + 1] = cy;
            new_xyz[(size_t)(b * NPOINT + it) * 3 + 2] = cz;
            out_newxyzT[(size_t)b * 3 * NPOINT + 0 * NPOINT + it] = cx;
            out_newxyzT[(size_t)b * 3 * NPOINT + 1 * NPOINT + it] = cy;
            out_newxyzT[(size_t)b * 3 * NPOINT + 2 * NPOINT + it] = cz;
        }
        float bv = -1.0f; int bi = 0;
        for (int i = tid; i < NPTS; i += 1024) {
            const float dx = xb[i] - cx;
            const float dy = xb[NPTS + i] - cy;
            const float dz = xb[2 * NPTS + i] - cz;
            const float d  = dx * dx + dy * dy + dz * dz;
            const float nd = fminf(dist[i], d);
            dist[i] = nd;
            if (nd > bv || (nd == bv && i < bi)) { bv = nd; bi = i; }
        }
        #pragma unroll
        for (int off = 16; off > 0; off >>= 1) {
            const float ov = __shfl_xor(bv, off, 32);
            const int   oi = __shfl_xor(bi, off, 32);
            if (ov > bv || (ov == bv && oi < bi)) { bv = ov; bi = oi; }
        }
        if (lane == 0) { sv[wv] = bv; si[wv] = bi; }
        __syncthreads();
        if (wv == 0) {
            bv = sv[lane]; bi = si[lane];
            #pragma unroll
            for (int off = 16; off > 0; off >>= 1) {
                const float ov = __shfl_xor(bv, off, 32);
                const int   oi = __shfl_xor(bi, off, 32);
                if (ov > bv || (ov == bv && oi < bi)) { bv = ov; bi = oi; }
            }
            if (lane == 0) sfar = bi;
        }
        __syncthreads();
        far = sfar;
    }
}

// ---------------------------------------------------------------------------
// 2. Ball query: one wave32 per center; ballot+popcount prefix collects the
//    first NSAMPLE ascending indices with d^2 <= r^2, fill with first.
// ---------------------------------------------------------------------------
__global__ __launch_bounds__(256) void qball_kernel(
    const float* __restrict__ xyz,
    const float* __restrict__ new_xyz,
    int* __restrict__ idx)
{
    const int cid  = (blockIdx.x * 256 + threadIdx.x) >> 5;
    const int lane = threadIdx.x & 31;
    const int w    = threadIdx.x >> 5;
    const int b    = cid >> 10;
    const float* xb = xyz + (size_t)b * 3 * NPTS;

    __shared__ int buf[8][NSAMPLE];

    const float cx = new_xyz[cid * 3 + 0];
    const float cy = new_xyz[cid * 3 + 1];
    const float cz = new_xyz[cid * 3 + 2];

    int cnt = 0;
    for (int base = 0; base < NPTS && cnt < NSAMPLE; base += 32) {
        const int i = base + lane;
        const float dx = xb[i] - cx;
        const float dy = xb[NPTS + i] - cy;
        const float dz = xb[2 * NPTS + i] - cz;
        const bool in = (dx * dx + dy * dy + dz * dz) <= RADIUS2;
        const unsigned bal = (unsigned)__ballot(in);
        const int pos = cnt + __popc(bal & ((1u << lane) - 1u));
        if (in && pos < NSAMPLE) buf[w][pos] = i;
        cnt += __popc(bal);
    }
    __builtin_amdgcn_wave_barrier();
    if (cnt > NSAMPLE) cnt = NSAMPLE;
    if (lane >= cnt) buf[w][lane] = buf[w][0];   // cnt >= 1 always (self)
    __builtin_amdgcn_wave_barrier();
    idx[(size_t)cid * NSAMPLE + lane] = buf[w][lane];
}

// ---------------------------------------------------------------------------
// 3. Feature build (rows: grouped then fps), stride 16, zero-pad 9..15.
// ---------------------------------------------------------------------------
__global__ __launch_bounds__(256) void build_kernel(
    const float* __restrict__ xyz, const float* __restrict__ points,
    const float* __restrict__ new_xyz, const int* __restrict__ idx,
    const int* __restrict__ fps_idx, float* __restrict__ X0)
{
    const int r = blockIdx.x * 256 + threadIdx.x;
    if (r >= TOT_ROWS) return;
    float f[16];
    #pragma unroll
    for (int j = 0; j < 16; ++j) f[j] = 0.0f;

    if (r < NP_ROWS) {
        const int center = r >> 5;
        const int b = center >> 10;
        const int i = idx[r];
        f[0] = xyz[(size_t)b * 3 * NPTS + i]            - new_xyz[center * 3 + 0];
        f[1] = xyz[(size_t)b * 3 * NPTS + NPTS + i]     - new_xyz[center * 3 + 1];
        f[2] = xyz[(size_t)b * 3 * NPTS + 2 * NPTS + i] - new_xyz[center * 3 + 2];
        #pragma unroll
        for (int c = 0; c < 6; ++c)
            f[3 + c] = points[(size_t)b * 6 * NPTS + (size_t)c * NPTS + i];
    } else {
        const int center = r - NP_ROWS;
        const int b = center >> 10;
        const int fi = fps_idx[center];
        f[0] = new_xyz[center * 3 + 0];
        f[1] = new_xyz[center * 3 + 1];
        f[2] = new_xyz[center * 3 + 2];
        #pragma unroll
        for (int c = 0; c < 6; ++c)
            f[3 + c] = points[(size_t)b * 6 * NPTS + (size_t)c * NPTS + fi];
    }
    #pragma unroll
    for (int j = 0; j < 16; ++j) X0[(size_t)r * 16 + j] = f[j];
}

// ---------------------------------------------------------------------------
// 3b. Pack a B-matrix (K x Cout logical) into WMMA B-fragment order:
//     out[((kc*2+h)*Cout + n)*2 + e] = B[k][n], k = kc*4 + 2h + e, 0 if k>=K.
//     src_k_major==1 : S is (K, Cout) row-major   (attention 'a' matrix)
//     src_k_major==0 : S is (Cout, K) row-major   (MLP weight W)
// ---------------------------------------------------------------------------
__global__ __launch_bounds__(256) void packb_kernel(
    const float* __restrict__ S, float* __restrict__ out,
    int K, int Cout, int kchunks, int src_k_major)
{
    const int t = blockIdx.x * 256 + threadIdx.x;
    const int total = kchunks * 2 * Cout * 2;
    if (t >= total) return;
    const int e  = t & 1;
    const int n  = (t >> 1) % Cout;
    const int kh = (t >> 1) / Cout;          // kc*2 + h
    const int k  = (kh >> 1) * 4 + (kh & 1) * 2 + e;
    float val = 0.0f;
    if (k < K)
        val = src_k_major ? S[(size_t)k * Cout + n] : S[(size_t)n * K + k];
    out[t] = val;
}

// ---------------------------------------------------------------------------
// 4. GEMM + BN + ReLU with V_WMMA_F32_16X16X4_F32, branch-free inner loop:
//    one float2 X load + one float2 packed-W load + wmma, fully unrolled.
// ---------------------------------------------------------------------------
template<int STRIDE, int KCH, int COUT>
__global__ __launch_bounds__(256) void mlp_wmma_kernel(
    const float* __restrict__ X,  const float* __restrict__ Wpk,
    const float* __restrict__ bb, const float* __restrict__ g,
    const float* __restrict__ be, const float* __restrict__ m,
    const float* __restrict__ v,  float* __restrict__ Y)
{
    const int lane = threadIdx.x & 31;
    const int half = lane >> 4;
    const int ln   = lane & 15;
    const int mt   = blockIdx.x * 8 + (threadIdx.x >> 5);
    const float* Xrow = X + (size_t)(mt * 16 + ln) * STRIDE;
    __builtin_prefetch(Xrow, 0, 1);   // global_prefetch_b8

    #pragma unroll
    for (int nt = 0; nt < (COUT >> 4); ++nt) {
        const int col = nt * 16 + ln;
        const float alpha = g[col] * rsqrtf(v[col] + BN_EPS);
        const float beta  = (bb[col] - m[col]) * alpha + be[col];
        v8f acc = {0.f, 0.f, 0.f, 0.f, 0.f, 0.f, 0.f, 0.f};
        #pragma unroll
        for (int kc = 0; kc < KCH; ++kc) {
            const float2 p = *(const float2*)(Xrow + kc * 4 + 2 * half);
            const float2 q = *(const float2*)(Wpk +
                               ((size_t)(kc * 2 + half) * COUT + col) * 2);
            v2f av = {p.x, p.y};
            v2f bv = {q.x, q.y};
            acc = __builtin_amdgcn_wmma_f32_16x16x4_f32(
                false, av, false, bv, (short)0, acc, false, false);
        }
        float* Yb = Y + (size_t)(mt * 16 + 8 * half) * COUT + col;
        #pragma unroll
        for (int vv = 0; vv < 8; ++vv) {
            const float y = acc[vv] * alpha + beta;
            Yb[(size_t)vv * COUT] = fmaxf(y, 0.0f);
        }
    }
}

// ---------------------------------------------------------------------------
// 5. Attention: per center (one wave). kc=0 chunk (touches delta_p) is peeled
//    and built with branch-free selects; kc>=1 chunks are pure f[j]-h[m][j]
//    (k=131 pad multiplies a pre-packed zero in B). Softmax over the 32
//    samples is done on the accumulator layout (8 regs + shfl_xor(16)).
// ---------------------------------------------------------------------------
__global__ __launch_bounds__(256) void attn_kernel(
    const float* __restrict__ xyz, const float* __restrict__ new_xyz,
    const int* __restrict__ idx,  const float* __restrict__ H,   // (TOT_ROWS,128)
    const float* __restrict__ Apk,                               // packed (33 ch)
    float* __restrict__ out)
{
    const int c    = blockIdx.x * 8 + (threadIdx.x >> 5);
    const int lane = threadIdx.x & 31;
    const int half = lane >> 4;
    const int ln   = lane & 15;
    const int b    = c >> 10;
    const int s    = c & 1023;
    const float* xb = xyz + (size_t)b * 3 * NPTS;
    const float* Hc = H + (size_t)c * 32 * 128;
    const float* fc = H + (size_t)(NP_ROWS + c) * 128;
    const float* H0 = Hc + (size_t)ln * 128;          // row ln
    const float* H1 = Hc + (size_t)(16 + ln) * 128;   // row 16+ln
    const float nx = new_xyz[c * 3 + 0];
    const float ny = new_xyz[c * 3 + 1];
    const float nz = new_xyz[c * 3 + 2];
    float* poolOut = out + (size_t)BATCH * 3 * NPOINT
                   + (size_t)b * 128 * NPOINT + s;

    // ---- hoisted kc==0 A-fragments (branch-free selects) ----
    const int i0 = idx[(size_t)c * 32 + ln];
    const int i1 = idx[(size_t)c * 32 + 16 + ln];
    const int koff = half ? 2 * NPTS : 0;             // k=0 or k=2
    const float nsel = half ? nz : nx;
    const float a0x = nsel - xb[koff + i0];
    const float a1x = nsel - xb[koff + i1];
    const float g0  = ny - xb[NPTS + i0];             // k=1 (half 0)
    const float g1  = ny - xb[NPTS + i1];
    const float h0  = fc[0] - H0[0];                  // k=3 (half 1)
    const float h1  = fc[0] - H1[0];
    const float a0y = half ? h0 : g0;
    const float a1y = half ? h1 : g1;
    const v2f a0c0 = {a0x, a0y};
    const v2f a1c0 = {a1x, a1y};

    for (int nt = 0; nt < 8; ++nt) {
        const int col = nt * 16 + ln;
        v8f acc0 = {0.f, 0.f, 0.f, 0.f, 0.f, 0.f, 0.f, 0.f};
        v8f acc1 = {0.f, 0.f, 0.f, 0.f, 0.f, 0.f, 0.f, 0.f};
        {   // kc == 0
            const float2 q = *(const float2*)(Apk + ((size_t)half * 128 + col) * 2);
            v2f bf = {q.x, q.y};
            acc0 = __builtin_amdgcn_wmma_f32_16x16x4_f32(
                false, a0c0, false, bf, (short)0, acc0, false, false);
            acc1 = __builtin_amdgcn_wmma_f32_16x16x4_f32(
                false, a1c0, false, bf, (short)0, acc1, false, false);
        }
        #pragma unroll 8
        for (int kc = 1; kc < 33; ++kc) {
            const int j = kc * 4 + 2 * half - 3;      // 1..127
            const float fj0 = fc[j], fj1 = fc[j + 1];
            v2f a0 = {fj0 - H0[j], fj1 - H0[j + 1]};
            v2f a1 = {fj0 - H1[j], fj1 - H1[j + 1]};
            const float2 q = *(const float2*)(Apk +
                               ((size_t)(kc * 2 + half) * 128 + col) * 2);
            v2f bf = {q.x, q.y};
            acc0 = __builtin_amdgcn_wmma_f32_16x16x4_f32(
                false, a0, false, bf, (short)0, acc0, false, false);
            acc1 = __builtin_amdgcn_wmma_f32_16x16x4_f32(
                false, a1, false, bf, (short)0, acc1, false, false);
        }
        // leaky relu
        #pragma unroll
        for (int vv = 0; vv < 8; ++vv) {
            acc0[vv] = (acc0[vv] > 0.f) ? acc0[vv] : LRELU_A * acc0[vv];
            acc1[vv] = (acc1[vv] > 0.f) ? acc1[vv] : LRELU_A * acc1[vv];
        }
        // column softmax over 32 rows + pooled sum
        float mx = -3.4e38f;
        #pragma unroll
        for (int vv = 0; vv < 8; ++vv) mx = fmaxf(mx, fmaxf(acc0[vv], acc1[vv]));
        mx = fmaxf(mx, __shfl_xor(mx, 16, 32));
        float ssum = 0.f, pool = 0.f;
        #pragma unroll
        for (int vv = 0; vv < 8; ++vv) {
            const float e0 = __expf(acc0[vv] - mx);   // row vv+8*half
            const float e1 = __expf(acc1[vv] - mx);   // row 16+vv+8*half
            ssum += e0 + e1;
            const float hh0 = Hc[(size_t)(vv + 8 * half) * 128 + col];
            const float hh1 = Hc[(size_t)(16 + vv + 8 * half) * 128 + col];
            pool += e0 * hh0 + e1 * hh1;
        }
        ssum += __shfl_xor(ssum, 16, 32);
        pool += __shfl_xor(pool, 16, 32);
        if (half == 0) poolOut[(size_t)col * NPOINT] = pool / ssum;
    }
}

// ---------------------------------------------------------------------------
// Launch: FPS -> qball -> build -> pack -> 3x WMMA GEMM -> attn.
// ---------------------------------------------------------------------------
extern "C" void kernel_launch(void* const* d_in, const int* in_sizes, int n_in,
                              void* d_out, int out_size, void* d_ws, size_t ws_size,
                              hipStream_t stream)
{
    (void)in_sizes; (void)n_in; (void)out_size; (void)ws_size;
    const float* xyz    = (const float*)d_in[0];
    const float* points = (const float*)d_in[1];
    const float *Wp[3], *bp[3], *gp[3], *bep[3], *mp[3], *vp[3];
    for (int l = 0; l < 3; ++l) {
        Wp[l]  = (const float*)d_in[2 + 6 * l];
        bp[l]  = (const float*)d_in[3 + 6 * l];
        gp[l]  = (const float*)d_in[4 + 6 * l];
        bep[l] = (const float*)d_in[5 + 6 * l];
        mp[l]  = (const float*)d_in[6 + 6 * l];
        vp[l]  = (const float*)d_in[7 + 6 * l];
    }
    const float* amat = (const float*)d_in[20];
    float* out = (float*)d_out;

    char* ws = (char*)d_ws;
    size_t off = 0;
    auto walloc = [&](size_t bytes) -> void* {
        void* p = ws + off;
        off = (off + bytes + 255) & ~((size_t)255);
        return p;
    };
    int*   fps_idx = (int*)  walloc((size_t)FPS_ROWS * 4);
    float* nxyz    = (float*)walloc((size_t)FPS_ROWS * 3 * 4);
    int*   idx     = (int*)  walloc((size_t)NP_ROWS * 4);
    float* wpk0    = (float*)walloc((size_t)4  * 2 * 64  * 2 * 4);
    float* wpk1    = (float*)walloc((size_t)16 * 2 * 64  * 2 * 4);
    float* wpk2    = (float*)walloc((size_t)16 * 2 * 128 * 2 * 4);
    float* apk     = (float*)walloc((size_t)33 * 2 * 128 * 2 * 4);
    float* P       = (float*)walloc((size_t)TOT_ROWS * 64 * 4);    // X0 & X2
    float* Q       = (float*)walloc((size_t)TOT_ROWS * 64 * 4);    // X1
    float* R       = (float*)walloc((size_t)TOT_ROWS * 128 * 4 + 256); // X3+slack

    fps_kernel<<<BATCH, 1024, 0, stream>>>(xyz, fps_idx, nxyz, out);
    qball_kernel<<<(FPS_ROWS * 32) / 256, 256, 0, stream>>>(xyz, nxyz, idx);
    build_kernel<<<(TOT_ROWS + 255) / 256, 256, 0, stream>>>(
        xyz, points, nxyz, idx, fps_idx, P);

    packb_kernel<<<(4  * 2 * 64  * 2 + 255) / 256, 256, 0, stream>>>(
        Wp[0], wpk0, 9,   64,  4,  0);
    packb_kernel<<<(16 * 2 * 64  * 2 + 255) / 256, 256, 0, stream>>>(
        Wp[1], wpk1, 64,  64,  16, 0);
    packb_kernel<<<(16 * 2 * 128 * 2 + 255) / 256, 256, 0, stream>>>(
        Wp[2], wpk2, 64,  128, 16, 0);
    packb_kernel<<<(33 * 2 * 128 * 2 + 255) / 256, 256, 0, stream>>>(
        amat,  apk,  131, 128, 33, 1);

    const int mtiles = TOT_ROWS / 16;   // 33792, divisible by 8
    mlp_wmma_kernel<16, 4, 64><<<mtiles / 8, 256, 0, stream>>>(
        P, wpk0, bp[0], gp[0], bep[0], mp[0], vp[0], Q);
    mlp_wmma_kernel<64, 16, 64><<<mtiles / 8, 256, 0, stream>>>(
        Q, wpk1, bp[1], gp[1], bep[1], mp[1], vp[1], P);
    mlp_wmma_kernel<64, 16, 128><<<mtiles / 8, 256, 0, stream>>>(
        P, wpk2, bp[2], gp[2], bep[2], mp[2], vp[2], R);

    attn_kernel<<<FPS_ROWS / 8, 256, 0, stream>>>(xyz, nxyz, idx, R, apk, out);
}